// UnitaryBranching_23691039604824
// MI455X (gfx1250) — compile-verified
//
#include <hip/hip_runtime.h>

#define BF   2
#define NH   8
#define DIM  64
#define PAD  (BF + 1)
#define MAXL 32

typedef __attribute__((ext_vector_type(2))) float v2f;
typedef __attribute__((ext_vector_type(8))) float v8f;

// Reciprocals for the Horner-Taylor recurrence (index by t), avoids per-iteration
// v_div_scale/v_rcp sequences in the hot loop.
__device__ __constant__ float kInvT[17] = {
    0.0f,        1.0f,        1.0f / 2.0f,  1.0f / 3.0f,
    1.0f / 4.0f, 1.0f / 5.0f, 1.0f / 6.0f,  1.0f / 7.0f,
    1.0f / 8.0f, 1.0f / 9.0f, 1.0f / 10.0f, 1.0f / 11.0f,
    1.0f / 12.0f, 1.0f / 13.0f, 1.0f / 14.0f, 1.0f / 15.0f,
    1.0f / 16.0f};

// One wave computes one 16x16 f32 tile of C = A(64x64) * B(64x64), both row-major
// in LDS, using V_WMMA_F32_16X16X4_F32 (K=4 per instruction, 16 instructions).
// ISA 7.12.2 layouts:
//   A 16x4 frag : m = lane&15, k = 2*(lane>>4) + v      (v = 0,1 -> a.x,a.y)
//   B 4x16 frag : n = lane&15, k = 2*(lane>>4) + v
//   C 16x16     : VGPR r -> row = r + 8*(lane>>4), col = lane&15
__device__ __forceinline__ v8f tile_mm64(const float* A, const float* B,
                                         int wr, int wc, int lane) {
    const int m    = lane & 15;
    const int half = lane >> 4;
    v8f acc = {};
#pragma unroll
    for (int k0 = 0; k0 < DIM; k0 += 4) {
        const float* ap = A + (wr * 16 + m) * DIM + k0 + 2 * half;
        const float* bp = B + (k0 + 2 * half) * DIM + wc * 16 + m;
        v2f a, b;
        a.x = ap[0];
        a.y = ap[1];
        b.x = bp[0];
        b.y = bp[DIM];
        acc = __builtin_amdgcn_wmma_f32_16x16x4_f32(
            /*neg_a=*/false, a, /*neg_b=*/false, b,
            /*c_mod=*/(short)0, acc, /*reuse_a=*/false, /*reuse_b=*/false);
    }
    return acc;
}

// ---------------------------------------------------------------------------
// Kernel 1: prims[i] = expm((A_i - A_i^T)/DIM), Horner-Taylor, 16 terms.
// grid = 17 blocks, 512 threads (16 wave32, one 16x16 tile each).
// Double-buffered S: one barrier per Horner step.
// ---------------------------------------------------------------------------
__global__ void __launch_bounds__(512)
expm_kernel(const float* __restrict__ raw, float* __restrict__ prims) {
    __shared__ float H[DIM * DIM];
    __shared__ float Sb[2][DIM * DIM];

    const int mat = blockIdx.x;
    const float* A = raw + (size_t)mat * DIM * DIM;
    const int tid = threadIdx.x;

    for (int i = tid; i < DIM * DIM; i += 512) {
        int r = i >> 6, c = i & 63;
        H[i]     = (A[r * DIM + c] - A[c * DIM + r]) * (1.0f / (float)DIM);
        Sb[0][i] = (r == c) ? 1.0f : 0.0f;
    }
    __syncthreads();

    const int wave = tid >> 5, lane = tid & 31;
    const int wr = wave >> 2, wc = wave & 3;
    const int n = lane & 15, half = lane >> 4;

    // S <- I + (H*S)/t  for t = 16..1  ==> S -> expm(H)
    int cur = 0;
    for (int t = 16; t >= 1; --t) {
        v8f c = tile_mm64(H, Sb[cur], wr, wc, lane);
        const float inv = kInvT[t];
        float* __restrict__ Sn = Sb[cur ^ 1];  // disjoint from read buffer
#pragma unroll
        for (int r = 0; r < 8; ++r) {
            int row = wr * 16 + r + 8 * half;
            int col = wc * 16 + n;
            Sn[row * DIM + col] = c[r] * inv + ((row == col) ? 1.0f : 0.0f);
        }
        __syncthreads();
        cur ^= 1;
    }

    // Vectorized writeback (global_store_b128)
    const float4* Sv = (const float4*)Sb[cur];
    float4* Pv = (float4*)(prims + (size_t)mat * DIM * DIM);
    for (int i = tid; i < (DIM * DIM) / 4; i += 512) Pv[i] = Sv[i];
}

// ---------------------------------------------------------------------------
// Kernel 2: maps[p,h] = init(p) * prod_d gate(w_d) ? branch[w_d,h] : I
// (faithful gate: only w in [0,BF) multiplies; init = sos_repr iff w_0 == -1)
// grid = (N, NH) blocks, 512 threads. Double-buffered chain product.
// ---------------------------------------------------------------------------
__global__ void __launch_bounds__(512)
maps_kernel(const float* __restrict__ prims, const int* __restrict__ pw,
            float* __restrict__ maps, int L) {
    __shared__ float Sb[2][DIM * DIM];
    __shared__ float B0s[DIM * DIM];
    __shared__ float B1s[DIM * DIM];

    const int p = blockIdx.x;
    const int h = blockIdx.y;
    const int tid = threadIdx.x;
    const int* path = pw + (size_t)p * L;

    const bool sos = (path[0] == -1);  // uniform across block
    const float4* sosV = (const float4*)(prims + (size_t)(BF * NH) * DIM * DIM);
    const float4* M0v  = (const float4*)(prims + (size_t)(0 * NH + h) * DIM * DIM);
    const float4* M1v  = (const float4*)(prims + (size_t)(1 * NH + h) * DIM * DIM);
    float4* S0v = (float4*)Sb[0];
    float4* B0v = (float4*)B0s;
    float4* B1v = (float4*)B1s;

    // Stage branch matrices + init product with b128 traffic.
    for (int i = tid; i < (DIM * DIM) / 4; i += 512) {
        B0v[i] = M0v[i];
        B1v[i] = M1v[i];
        if (sos) {
            S0v[i] = sosV[i];
        } else {
            int base = i * 4;              // flat element index of .x
            int r = base >> 6;
            float4 v;
            v.x = (r == (base & 63))       ? 1.0f : 0.0f;
            v.y = (r == ((base + 1) & 63)) ? 1.0f : 0.0f;
            v.z = (r == ((base + 2) & 63)) ? 1.0f : 0.0f;
            v.w = (r == ((base + 3) & 63)) ? 1.0f : 0.0f;
            S0v[i] = v;
        }
    }
    __syncthreads();

    const int wave = tid >> 5, lane = tid & 31;
    const int wr = wave >> 2, wc = wave & 3;
    const int n = lane & 15, half = lane >> 4;

    int cur = 0;
    for (int d = 0; d < L; ++d) {
        const int w = path[d];           // uniform (scalar) load
        if (w >= 0 && w < BF) {          // uniform branch: EXEC stays all-1s
            const float* Bm = (w == 0) ? B0s : B1s;
            v8f c = tile_mm64(Sb[cur], Bm, wr, wc, lane);
            float* __restrict__ Sn = Sb[cur ^ 1];
#pragma unroll
            for (int r = 0; r < 8; ++r)
                Sn[(wr * 16 + r + 8 * half) * DIM + wc * 16 + n] = c[r];
            __syncthreads();
            cur ^= 1;
        }
    }

    // 256 MB total output: b128 stores, fully coalesced.
    const float4* Sv = (const float4*)Sb[cur];
    float4* out = (float4*)(maps + ((size_t)p * NH + h) * DIM * DIM);
    for (int i = tid; i < (DIM * DIM) / 4; i += 512) out[i] = Sv[i];
}

// ---------------------------------------------------------------------------
// Kernel 3: steps[i,j] = max(len_i, len_j) - common_prefix(i,j)
// 16x16 output tile per 256-thread block, path rows staged in LDS.
// ---------------------------------------------------------------------------
__global__ void __launch_bounds__(256)
steps_kernel(const int* __restrict__ pw, float* __restrict__ out, int N, int L) {
    __shared__ int Ri[16 * MAXL];
    __shared__ int Rj[16 * MAXL];
    __shared__ int Li[16];
    __shared__ int Lj[16];

    const int bi = blockIdx.y * 16;
    const int bj = blockIdx.x * 16;
    const int tid = threadIdx.x;

    for (int i = tid; i < 16 * L; i += 256) {
        int r = i / L, c = i % L;
        Ri[r * MAXL + c] = pw[(size_t)(bi + r) * L + c];
        Rj[r * MAXL + c] = pw[(size_t)(bj + r) * L + c];
    }
    __syncthreads();

    if (tid < 16) {
        int cnt = 0;
        for (int c = 0; c < L; ++c) cnt += (Ri[tid * MAXL + c] != PAD);
        Li[tid] = cnt;
    } else if (tid < 32) {
        int t = tid - 16, cnt = 0;
        for (int c = 0; c < L; ++c) cnt += (Rj[t * MAXL + c] != PAD);
        Lj[t] = cnt;
    }
    __syncthreads();

    const int li = tid >> 4, lj = tid & 15;
    int cp = 0;
    bool run = true;
    for (int c = 0; c < L; ++c) {
        int a = Ri[li * MAXL + c];
        int b = Rj[lj * MAXL + c];
        run = run && (a == b);
        cp += (run && a != PAD && b != PAD) ? 1 : 0;
    }
    int mx = Li[li] > Lj[lj] ? Li[li] : Lj[lj];
    out[(size_t)(bi + li) * N + (bj + lj)] = (float)(mx - cp);
}

// ---------------------------------------------------------------------------
extern "C" void kernel_launch(void* const* d_in, const int* in_sizes, int n_in,
                              void* d_out, int out_size, void* d_ws, size_t ws_size,
                              hipStream_t stream) {
    const float* raw = (const float*)d_in[0];   // [BF*NH+1, 64, 64] f32
    const int*   pw  = (const int*)d_in[1];     // [N, L] i32
    float* out = (float*)d_out;

    const int n_prims = in_sizes[0] / (DIM * DIM);  // 17
    const int N = 2048;
    const int L = in_sizes[1] / N;                  // ~11

    float* prims = (float*)d_ws;                    // 17*4096 f32 scratch

    expm_kernel<<<dim3(n_prims), 512, 0, stream>>>(raw, prims);
    maps_kernel<<<dim3(N, NH), 512, 0, stream>>>(prims, pw, out, L);

    float* steps_out = out + (size_t)N * NH * DIM * DIM;
    steps_kernel<<<dim3(N / 16, N / 16), 256, 0, stream>>>(pw, steps_out, N, L);
}